// GraphPool_2018634629399
// MI455X (gfx1250) — compile-verified
//
#include <hip/hip_runtime.h>

// GraphPool: out[s,a,:] = max over {self, valid neighbors} of atoms[s,nb,:],
// zeroed where degree==0.  S=512 molecules, A=128 atoms, F=128 feats, D=6.
#define S_DIM 512
#define A_DIM 128
#define F_DIM 128
#define D_DIM 6
#define BLOCK 256
#define F4    (F_DIM / 4)        // 32 float4 per atom row
#define ELEMS (A_DIM * F4)       // 4096 float4 per molecule

#if __has_builtin(__builtin_amdgcn_global_load_async_to_lds_b128) && \
    __has_builtin(__builtin_amdgcn_s_wait_asynccnt)
#define USE_ASYNC_LDS 1
#else
#define USE_ASYNC_LDS 0
#endif

typedef __attribute__((ext_vector_type(4))) int v4i;
typedef __attribute__((address_space(1))) v4i gv4i;   // global (AS1) int4
typedef __attribute__((address_space(3))) v4i lv4i;   // LDS    (AS3) int4

__global__ __launch_bounds__(BLOCK) void graphpool_max_kernel(
    const float* __restrict__ atoms,
    const int* __restrict__ edges,
    float* __restrict__ out)
{
    __shared__ float lds_atoms[A_DIM * F_DIM];   // 64 KB: one molecule's atoms
    __shared__ int   lds_edge[A_DIM * D_DIM];    // 3 KB: one molecule's edges

    const int s   = blockIdx.x;
    const int tid = threadIdx.x;

    const float4* __restrict__ gatoms =
        (const float4*)(atoms + (size_t)s * A_DIM * F_DIM);

    // ---- Stage atoms[s] (128x128 f32 = 64KB) into LDS -------------------
#if USE_ASYNC_LDS
    // CDNA5 async copy: global -> LDS directly, tracked by ASYNCcnt.
    for (int i = tid; i < ELEMS; i += BLOCK) {
        __builtin_amdgcn_global_load_async_to_lds_b128(
            (gv4i*)(gatoms + i),
            (lv4i*)(lds_atoms + 4 * i),
            0, 0);
    }
#else
    float4* lat4 = (float4*)lds_atoms;
    for (int i = tid; i < ELEMS; i += BLOCK) lat4[i] = gatoms[i];
#endif

    // ---- Stage edges[s] (128x6 i32) into LDS ----------------------------
    const int* __restrict__ gedges = edges + (size_t)s * A_DIM * D_DIM;
    for (int i = tid; i < A_DIM * D_DIM; i += BLOCK) lds_edge[i] = gedges[i];

#if USE_ASYNC_LDS
    __builtin_amdgcn_s_wait_asynccnt(0);
#endif
    __syncthreads();

    // ---- Gather + max ----------------------------------------------------
    float4* __restrict__ gout = (float4*)(out + (size_t)s * A_DIM * F_DIM);
    const float4* latoms = (const float4*)lds_atoms;

    for (int i = tid; i < ELEMS; i += BLOCK) {
        const int a  = i >> 5;   // atom index: uniform across a 32-lane group
        const int fc = i & 31;   // float4 chunk within the feature row

        float4 m = latoms[a * F4 + fc];   // self row (always included)
        int deg = 0;
#pragma unroll
        for (int d = 0; d < D_DIM; ++d) {
            const int nb = lds_edge[a * D_DIM + d];  // wave-uniform broadcast
            if (nb >= 0) {                            // wave-uniform branch
                ++deg;
                const float4 v = latoms[nb * F4 + fc];
                m.x = fmaxf(m.x, v.x);
                m.y = fmaxf(m.y, v.y);
                m.z = fmaxf(m.z, v.z);
                m.w = fmaxf(m.w, v.w);
            }
        }
        if (deg == 0) m = make_float4(0.f, 0.f, 0.f, 0.f);
        gout[i] = m;
    }
}

extern "C" void kernel_launch(void* const* d_in, const int* in_sizes, int n_in,
                              void* d_out, int out_size, void* d_ws, size_t ws_size,
                              hipStream_t stream) {
    (void)in_sizes; (void)n_in; (void)out_size; (void)d_ws; (void)ws_size;
    const float* atoms = (const float*)d_in[0];
    const int*   edges = (const int*)d_in[1];
    float*       out   = (float*)d_out;
    graphpool_max_kernel<<<S_DIM, BLOCK, 0, stream>>>(atoms, edges, out);
}